// Model_18021682774325
// MI455X (gfx1250) — compile-verified
//
#include <hip/hip_runtime.h>
#include <hip/hip_bf16.h>

// Reference collapses to one RNN step (only out[0] is used):
//   h   = tanh(x[0]*W_ih^T + b_ih + b_hh)   (BATCH,5)   -- W_hh is dead
//   out = relu(h*W_fc^T + b_fc)             (BATCH,3)
// Memory-latency bound (~128KB total traffic). FC done with exact-precision
// fp32 WMMA: V_WMMA_F32_16X16X4_F32, H=5 padded to 8 via two chained K=4 ops.

typedef __attribute__((ext_vector_type(2))) float v2f;
typedef __attribute__((ext_vector_type(8))) float v8f;

#define RNN_BATCH 8192
#define RNN_H 5
#define RNN_C 3

__global__ __launch_bounds__(256) void rnn_first_step_wmma(
    const float* __restrict__ x,     // (SEQ, BATCH, 1): only first BATCH floats used
    const float* __restrict__ W_ih,  // (5,1)
    const float* __restrict__ b_ih,  // (5,)
    const float* __restrict__ b_hh,  // (5,)
    const float* __restrict__ W_fc,  // (3,5) row-major
    const float* __restrict__ b_fc,  // (3,)
    float* __restrict__ out)         // (BATCH, 3)
{
    const int lane = threadIdx.x & 31;
    const int wave = threadIdx.x >> 5;
    const int tile = (blockIdx.x * 8 + wave) * 16;  // 16 batch rows per wave
    const int n2   = lane & 15;                     // N index (C/D, B) / M index (A)
    const bool hi  = lane >= 16;                    // lane-half: K={2,3} vs K={0,1}

    // Uniform weights/biases (scalar loads).
    const float w0 = W_ih[0], w1 = W_ih[1], w2 = W_ih[2], w3 = W_ih[3], w4 = W_ih[4];
    const float s0 = b_ih[0] + b_hh[0];
    const float s1 = b_ih[1] + b_hh[1];
    const float s2 = b_ih[2] + b_hh[2];
    const float s3 = b_ih[3] + b_hh[3];
    const float s4 = b_ih[4] + b_hh[4];

    // x[0, tile+n2, 0]; lanes 16-31 re-read the same cacheline (M index = n2).
    const float xv = x[tile + n2];

    // A-matrix (16x4 fp32): lane holds its two K-slices of row M = tile+n2.
    const float hj0 = tanhf(fmaf(xv, hi ? w2 : w0, hi ? s2 : s0));
    const float hj1 = tanhf(fmaf(xv, hi ? w3 : w1, hi ? s3 : s1));
    const float hj4 = tanhf(fmaf(xv, w4, s4));  // K=4 of padded second tile

    v2f a0; a0.x = hj0;              a0.y = hj1;
    v2f a1; a1.x = hi ? 0.0f : hj4;  a1.y = 0.0f;   // K=5,6,7 are zero padding

    // B-matrix (4x16 fp32): B[k][n] = W_fc[n][k] for n<3, k<5; zero otherwise.
    v2f b0; b0.x = 0.0f; b0.y = 0.0f;
    v2f b1; b1.x = 0.0f; b1.y = 0.0f;
    if (n2 < RNN_C) {
        const float* wr = W_fc + n2 * RNN_H;
        b0.x = hi ? wr[2] : wr[0];
        b0.y = hi ? wr[3] : wr[1];
        b1.x = hi ? 0.0f  : wr[4];
    }

    // C seeded with bias, broadcast down M (all 8 VGPRs identical per lane).
    const float cv = (n2 < RNN_C) ? b_fc[n2] : 0.0f;
    v8f c = {cv, cv, cv, cv, cv, cv, cv, cv};

    // D = A0*B0 + C ; D = A1*B1 + D   (K = 0..3 then padded K = 4..7)
    // EXEC is all-ones here (divergent paths above have re-converged).
    v8f d = __builtin_amdgcn_wmma_f32_16x16x4_f32(
        false, a0, false, b0, (short)0, c, false, false);
    d = __builtin_amdgcn_wmma_f32_16x16x4_f32(
        false, a1, false, b1, (short)0, d, false, false);

    // D layout: VGPR i -> M = i (lanes 0-15) / M = 8+i (lanes 16-31), N = n2.
    if (n2 < RNN_C) {
#pragma unroll
        for (int i = 0; i < 8; ++i) {
            const int m = tile + (hi ? 8 + i : i);
            out[m * RNN_C + n2] = fmaxf(d[i], 0.0f);
        }
    }
}

extern "C" void kernel_launch(void* const* d_in, const int* in_sizes, int n_in,
                              void* d_out, int out_size, void* d_ws, size_t ws_size,
                              hipStream_t stream) {
    (void)in_sizes; (void)n_in; (void)d_ws; (void)ws_size; (void)out_size;
    const float* x    = (const float*)d_in[0];
    const float* W_ih = (const float*)d_in[1];
    // d_in[2] = W_hh: dead (only out[0] of the scan is consumed; h0 == 0)
    const float* b_ih = (const float*)d_in[3];
    const float* b_hh = (const float*)d_in[4];
    const float* W_fc = (const float*)d_in[5];
    const float* b_fc = (const float*)d_in[6];
    float* out = (float*)d_out;

    // 8192 rows / (16 rows/wave * 8 waves/block) = 64 blocks of 256 threads.
    const int blocks = RNN_BATCH / (16 * 8);
    rnn_first_step_wmma<<<blocks, 256, 0, stream>>>(x, W_ih, b_ih, b_hh, W_fc, b_fc, out);
}